// GIN_32066225832508
// MI455X (gfx1250) — compile-verified
//
#include <hip/hip_runtime.h>
#include <hip/hip_bf16.h>

typedef __attribute__((ext_vector_type(2))) float v2f;
typedef __attribute__((ext_vector_type(8))) float v8f;

#define DCH 256
#define BN_EPS 1e-5f

// ---------------- utility: zero a small float buffer ----------------
__global__ void gin_zero_f32(float* p, int n) {
    int i = blockIdx.x * blockDim.x + threadIdx.x;
    if (i < n) p[i] = 0.0f;
}

// ---------------- per-channel sum & sum-of-squares ----------------
// X is N x 256 row-major; stats[c] = sum, stats[256+c] = sumsq.
__global__ void gin_col_stats(const float* __restrict__ X,
                              float* __restrict__ stats, int N) {
    int c = threadIdx.x;
    float s = 0.0f, s2 = 0.0f;
    for (int r = blockIdx.x; r < N; r += gridDim.x) {
        float v = X[r * DCH + c];
        s += v;
        s2 += v * v;
    }
    unsafeAtomicAdd(&stats[c], s);
    unsafeAtomicAdd(&stats[DCH + c], s2);
}

// ---------------- BatchNorm apply (optionally ReLU, optionally dual-store) --
__global__ void gin_bn_apply(const float* __restrict__ X,
                             const float* __restrict__ stats,
                             const float* __restrict__ g,
                             const float* __restrict__ b,
                             float* __restrict__ outA,
                             float* __restrict__ outB,
                             int N, int do_relu) {
    int idx = blockIdx.x * blockDim.x + threadIdx.x;
    if (idx >= N * DCH) return;
    int c = idx & (DCH - 1);
    float inv = 1.0f / (float)N;
    float mu  = stats[c] * inv;
    float var = stats[DCH + c] * inv - mu * mu;   // biased var (jnp.var)
    float h = (X[idx] - mu) * rsqrtf(var + BN_EPS) * g[c] + b[c];
    if (do_relu) h = fmaxf(h, 0.0f);
    outA[idx] = h;
    if (outB) outB[idx] = h;
}

// ---------------- edge scatter-add: acc[dst] += feat[src] ----------------
// One wave32 per edge; float4 gathers, native non-returning f32 atomics.
__global__ void gin_scatter_add(const float* __restrict__ feat,
                                float* __restrict__ acc,
                                const long long* __restrict__ edges, int E) {
    int e = blockIdx.x * (blockDim.x >> 5) + (threadIdx.x >> 5);
    if (e >= E) return;
    int lane = threadIdx.x & 31;
    long long s = edges[e];                  // src node
    long long d = edges[(long long)E + e];   // dst node
    const float4* sp = (const float4*)(feat + s * DCH);
    float* dp = acc + d * DCH;
    float4 v0 = sp[lane];
    float4 v1 = sp[lane + 32];
    int o = 4 * lane;
    unsafeAtomicAdd(dp + o + 0, v0.x);
    unsafeAtomicAdd(dp + o + 1, v0.y);
    unsafeAtomicAdd(dp + o + 2, v0.z);
    unsafeAtomicAdd(dp + o + 3, v0.w);
    unsafeAtomicAdd(dp + 128 + o + 0, v1.x);
    unsafeAtomicAdd(dp + 128 + o + 1, v1.y);
    unsafeAtomicAdd(dp + 128 + o + 2, v1.z);
    unsafeAtomicAdd(dp + 128 + o + 3, v1.w);
}

// ---------------- repack W (256x256 row-major) into WMMA B-frag layout -----
// Wp float2 index: ((nblk*64 + k4)*32 + lane)*4 + t
//   lane = (half<<4)|lr ; value = (W[k4*4+2*half][n], W[k4*4+2*half+1][n])
//   n = nblk*64 + t*16 + lr
// In the GEMM loop a wave then fetches all 4 B-frags of a k-step with two
// contiguous b128 loads (1KB per wave, perfectly coalesced).
__global__ void gin_repack_w(const float* __restrict__ W,
                             float* __restrict__ Wp) {
    int i = blockIdx.x * blockDim.x + threadIdx.x;   // 32768 threads
    int t    = i & 3;
    int lane = (i >> 2) & 31;
    int k4   = (i >> 7) & 63;
    int nblk = i >> 13;
    int half = lane >> 4, lr = lane & 15;
    int n = nblk * 64 + t * 16 + lr;
    int k = k4 * 4 + 2 * half;
    float2 v = make_float2(W[k * DCH + n], W[(k + 1) * DCH + n]);
    ((float2*)Wp)[i] = v;
}

// ---------------- fp32 WMMA GEMM: out = A(Nx256) @ W(256x256) + bias -------
// Block = 256 threads = 8 waves stacked in M (128-row block tile).
// Each wave owns a 16x64 strip (4 accumulators). Per k-step:
//   1 x b64 (A frag) + 2 x b128 (4 repacked B frags) + 4 x wmma.
// V_WMMA_F32_16X16X4_F32 layouts (ISA 7.12.2):
//   A 16x4:  lanes0-15 -> M=lane, K=0(v0),1(v1); lanes16-31 -> K=2,3
//   B 4x16:  lanes0-15 -> K=0(v0),1(v1); lanes16-31 -> K=2(v0),3(v1); N=lane&15
//   C 16x16: vgpr v -> M=v (lanes0-15) / M=v+8 (lanes16-31); N=lane&15
__global__ __launch_bounds__(256) void gin_gemm_bias(
        const float* __restrict__ A, const float* __restrict__ Wp,
        const float* __restrict__ bias, float* __restrict__ out,
        float* __restrict__ out2, int N, int do_relu) {
    int wave = threadIdx.x >> 5;
    int lane = threadIdx.x & 31;
    int half = lane >> 4;
    int lr   = lane & 15;
    int m0 = blockIdx.x * 128 + wave * 16;
    int n0 = blockIdx.y * 64;

    v8f acc0 = {}, acc1 = {}, acc2 = {}, acc3 = {};

    int arow = m0 + lr;
    if (arow >= N) arow = N - 1;               // clamp (stores are guarded)
    const float* aptr = A + (size_t)arow * DCH + 2 * half;
    // float4 view of repacked W: region stride per nblk = 4096 float4
    const float4* bptr = (const float4*)Wp
                         + (size_t)blockIdx.y * 4096 + lane * 2;

#pragma unroll 8
    for (int k4 = 0; k4 < 64; ++k4) {
        v2f a = *(const v2f*)(aptr + 4 * k4);  // K = 4*k4 + 2*half, +1
        float4 q01 = bptr[0];                  // B frags tiles 0,1
        float4 q23 = bptr[1];                  // B frags tiles 2,3
        bptr += 64;
        v2f b0 = {q01.x, q01.y};
        v2f b1 = {q01.z, q01.w};
        v2f b2 = {q23.x, q23.y};
        v2f b3 = {q23.z, q23.w};
        acc0 = __builtin_amdgcn_wmma_f32_16x16x4_f32(false, a, false, b0,
                                                     (short)0, acc0, false, false);
        acc1 = __builtin_amdgcn_wmma_f32_16x16x4_f32(false, a, false, b1,
                                                     (short)0, acc1, false, false);
        acc2 = __builtin_amdgcn_wmma_f32_16x16x4_f32(false, a, false, b2,
                                                     (short)0, acc2, false, false);
        acc3 = __builtin_amdgcn_wmma_f32_16x16x4_f32(false, a, false, b3,
                                                     (short)0, acc3, false, false);
    }

#define GIN_EPILOGUE(ACC, T)                                                  \
    {                                                                         \
        int n = n0 + 16 * (T) + lr;                                           \
        float bv = bias[n];                                                   \
        _Pragma("unroll")                                                     \
        for (int v = 0; v < 8; ++v) {                                         \
            int row = m0 + v + 8 * half;                                      \
            if (row < N) {                                                    \
                float val = (ACC)[v] + bv;                                    \
                if (do_relu) val = fmaxf(val, 0.0f);                          \
                out[(size_t)row * DCH + n] = val;                             \
                if (out2) out2[(size_t)row * DCH + n] = val;                  \
            }                                                                 \
        }                                                                     \
    }
    GIN_EPILOGUE(acc0, 0)
    GIN_EPILOGUE(acc1, 1)
    GIN_EPILOGUE(acc2, 2)
    GIN_EPILOGUE(acc3, 3)
#undef GIN_EPILOGUE
}

extern "C" void kernel_launch(void* const* d_in, const int* in_sizes, int n_in,
                              void* d_out, int out_size, void* d_ws, size_t ws_size,
                              hipStream_t stream) {
    const float*     x    = (const float*)d_in[0];
    const long long* edge = (const long long*)d_in[1];   // int64 (2, E) flat
    const float* g0  = (const float*)d_in[2];
    const float* b0  = (const float*)d_in[3];
    const float* W1a = (const float*)d_in[4];
    const float* b1a = (const float*)d_in[5];
    const float* g1  = (const float*)d_in[6];
    const float* bt1 = (const float*)d_in[7];
    const float* W1b = (const float*)d_in[8];
    const float* b1b = (const float*)d_in[9];
    const float* W2a = (const float*)d_in[10];
    const float* b2a = (const float*)d_in[11];
    const float* g2  = (const float*)d_in[12];
    const float* bt2 = (const float*)d_in[13];
    const float* W2b = (const float*)d_in[14];
    const float* b2b = (const float*)d_in[15];

    int N = in_sizes[0] / DCH;   // 100000
    int E = in_sizes[1] / 2;     // 1600000

    size_t nd = (size_t)N * DCH;
    float* bufA = (float*)d_ws;
    float* bufB = bufA + nd;
    float* bufC = bufB + nd;
    float* st0  = bufC + nd;          // [sum, sumsq] x 3 BNs
    float* st1  = st0 + 2 * DCH;
    float* st2  = st1 + 2 * DCH;
    float* Wp1a = st2 + 2 * DCH;      // 4 repacked weights, 64K floats each
    float* Wp1b = Wp1a + DCH * DCH;
    float* Wp2a = Wp1b + DCH * DCH;
    float* Wp2b = Wp2a + DCH * DCH;

    dim3 blk(256);
    int  nd_blocks   = (int)((nd + 255) / 256);
    int  edge_blocks = (E + 7) / 8;
    dim3 ggrid((N + 127) / 128, DCH / 64);

    // Repack all weights into WMMA B-fragment layout (cheap, 256KB each)
    gin_repack_w<<<128, blk, 0, stream>>>(W1a, Wp1a);
    gin_repack_w<<<128, blk, 0, stream>>>(W1b, Wp1b);
    gin_repack_w<<<128, blk, 0, stream>>>(W2a, Wp2a);
    gin_repack_w<<<128, blk, 0, stream>>>(W2b, Wp2b);

    // Fresh BN(256) on input
    gin_zero_f32<<<6, blk, 0, stream>>>(st0, 6 * DCH);
    gin_col_stats<<<1024, blk, 0, stream>>>(x, st0, N);
    gin_bn_apply<<<nd_blocks, blk, 0, stream>>>(x, st0, g0, b0,
                                                bufA, bufB, N, 0);
    // --- layer 1: h = h0 + segment_sum(h0[src] -> dst) ---
    gin_scatter_add<<<edge_blocks, blk, 0, stream>>>(bufA, bufB, edge, E);
    // z = h @ W1a + b1a
    gin_gemm_bias<<<ggrid, blk, 0, stream>>>(bufB, Wp1a, b1a, bufC, nullptr, N, 0);
    // h1 = relu(BN(z))
    gin_col_stats<<<1024, blk, 0, stream>>>(bufC, st1, N);
    gin_bn_apply<<<nd_blocks, blk, 0, stream>>>(bufC, st1, g1, bt1,
                                                bufA, nullptr, N, 1);
    // out1 = relu(h1 @ W1b + b1b); dual-store: bufB = gather src, bufC = acc
    gin_gemm_bias<<<ggrid, blk, 0, stream>>>(bufA, Wp1b, b1b, bufB, bufC, N, 1);
    // --- layer 2 ---
    gin_scatter_add<<<edge_blocks, blk, 0, stream>>>(bufB, bufC, edge, E);
    gin_gemm_bias<<<ggrid, blk, 0, stream>>>(bufC, Wp2a, b2a, bufA, nullptr, N, 0);
    gin_col_stats<<<1024, blk, 0, stream>>>(bufA, st2, N);
    gin_bn_apply<<<nd_blocks, blk, 0, stream>>>(bufA, st2, g2, bt2,
                                                bufB, nullptr, N, 1);
    gin_gemm_bias<<<ggrid, blk, 0, stream>>>(bufB, Wp2b, b2b,
                                             (float*)d_out, nullptr, N, 1);
}